// DSL_19791209300129
// MI455X (gfx1250) — compile-verified
//
#include <hip/hip_runtime.h>

typedef __attribute__((ext_vector_type(2))) float v2f;
typedef __attribute__((ext_vector_type(8))) float v8f;

#define NN 8192
#define DD 512
#define HH 256
#define KNN 8

__device__ __forceinline__ v8f wmma4(v2f a, v2f b, v8f c) {
  // emits v_wmma_f32_16x16x4_f32 (codegen-confirmed round 1)
  return __builtin_amdgcn_wmma_f32_16x16x4_f32(false, a, false, b, (short)0, c,
                                               false, false);
}

__device__ __forceinline__ float leaky(float v) {
  return v >= 0.0f ? v : 0.01f * v;
}

// Stage a 16 x 512 float tile (global row stride DD) into LDS with padded row
// stride 516 floats, using CDNA5 async global->LDS b128 copies (ASYNCcnt).
// LDS byte address = low 32 bits of the flat shared-aperture pointer
// (ISA 10.2).
__device__ __forceinline__ void stage_tile_async(const float* __restrict__ g,
                                                 const float* lds_ptr,
                                                 int tid) {
  const unsigned base = (unsigned)(uintptr_t)lds_ptr;
  for (int g4 = tid; g4 < 16 * (DD / 4); g4 += 256) {
    const int r = g4 >> 7;            // row 0..15
    const int c = (g4 & 127) << 2;    // col 0..508, float4 aligned
    const unsigned lds = base + (unsigned)(r * 516 + c) * 4u;
    const float* gp = g + (size_t)r * DD + c;
    asm volatile("global_load_async_to_lds_b128 %0, %1, off"
                 :
                 : "v"(lds), "v"(gp)
                 : "memory");
  }
  asm volatile("s_wait_asynccnt 0x0" ::: "memory");
}

// ---------------- Kernel 1: fused MLP  out = leaky(leaky(x@W1+b1)@W2+b2) ----
__global__ __launch_bounds__(256) void mlp_kernel(
    const float* __restrict__ x, const float* __restrict__ W1,
    const float* __restrict__ b1, const float* __restrict__ W2,
    const float* __restrict__ b2, float* __restrict__ out) {
  __shared__ float xs[16 * 516];   // x tile, padded stride (516 % 64 == 4)
  __shared__ float hs[16 * 260];   // hidden tile, padded stride
  const int tid = threadIdx.x;
  const int wave = tid >> 5, lane = tid & 31;
  const int half = lane >> 4, l16 = lane & 15;
  const int m0 = blockIdx.x * 16;

  stage_tile_async(x + (size_t)m0 * DD, xs, tid);
  __syncthreads();

  // Layer 1: h[16,256]
  for (int nt = wave; nt < HH / 16; nt += 8) {
    const int n0 = nt * 16;
    v8f acc = {};
    for (int k0 = 0; k0 < DD; k0 += 4) {
      v2f a, b;
      a.x = xs[l16 * 516 + k0 + 2 * half];
      a.y = xs[l16 * 516 + k0 + 2 * half + 1];
      b.x = W1[(size_t)(k0 + 2 * half) * HH + n0 + l16];
      b.y = W1[(size_t)(k0 + 2 * half + 1) * HH + n0 + l16];
      acc = wmma4(a, b, acc);
    }
    const float bias = b1[n0 + l16];
    for (int r = 0; r < 8; ++r) {
      int row = half * 8 + r;
      hs[row * 260 + n0 + l16] = leaky(acc[r] + bias);
    }
  }
  __syncthreads();

  // Layer 2: out[16,512]
  for (int nt = wave; nt < DD / 16; nt += 8) {
    const int n0 = nt * 16;
    v8f acc = {};
    for (int k0 = 0; k0 < HH; k0 += 4) {
      v2f a, b;
      a.x = hs[l16 * 260 + k0 + 2 * half];
      a.y = hs[l16 * 260 + k0 + 2 * half + 1];
      b.x = W2[(size_t)(k0 + 2 * half) * DD + n0 + l16];
      b.y = W2[(size_t)(k0 + 2 * half + 1) * DD + n0 + l16];
      acc = wmma4(a, b, acc);
    }
    const float bias = b2[n0 + l16];
    for (int r = 0; r < 8; ++r) {
      int row = half * 8 + r;
      out[(size_t)(m0 + row) * DD + n0 + l16] = leaky(acc[r] + bias);
    }
  }
}

// ---------------- Kernel 2: row L2-normalize into workspace ----------------
__global__ __launch_bounds__(256) void norm_kernel(const float* __restrict__ out,
                                                   float* __restrict__ xn) {
  __shared__ float red[256];
  const int row = blockIdx.x;
  float s = 0.0f;
  for (int c = threadIdx.x; c < DD; c += 256) {
    float v = out[(size_t)row * DD + c];
    s += v * v;
  }
  red[threadIdx.x] = s;
  __syncthreads();
  for (int off = 128; off > 0; off >>= 1) {
    if (threadIdx.x < off) red[threadIdx.x] += red[threadIdx.x + off];
    __syncthreads();
  }
  const float inv = 1.0f / (sqrtf(red[0]) + 1e-12f);
  for (int c = threadIdx.x; c < DD; c += 256)
    xn[(size_t)row * DD + c] = out[(size_t)row * DD + c] * inv;
}

// ------- Kernel 3: fused sim GEMM (xn@xn.T) + per-row top-8, no sim matrix --
__global__ __launch_bounds__(256) void knn_kernel(const float* __restrict__ xn,
                                                  float* __restrict__ dout) {
  __shared__ float qa[16 * 516];   // 16 query rows of xn
  __shared__ float sims[16 * 132]; // 16 x 128 tile of similarities (padded)
  const int tid = threadIdx.x;
  const int wave = tid >> 5, lane = tid & 31;
  const int half = lane >> 4, l16 = lane & 15;
  const int q0 = blockIdx.x * 16;

  stage_tile_async(xn + (size_t)q0 * DD, qa, tid);
  __syncthreads();

  float bestv[KNN];
  int besti[KNN];
  if (tid < 16) {
    for (int j = 0; j < KNN; ++j) { bestv[j] = -3.4e38f; besti[j] = 0; }
  }

  for (int c0 = 0; c0 < NN; c0 += 128) {
    const int ct = c0 + wave * 16;
    const float* bptr = xn + (size_t)(ct + l16) * DD;
    // prefetch next column-block's row for this lane (L2-resident anyway)
    {
      const int cn = (c0 + 128 < NN ? c0 + 128 : 0) + wave * 16;
      __builtin_prefetch(xn + (size_t)(cn + l16) * DD, 0, 3);
    }
    v8f acc = {};
    for (int k0 = 0; k0 < DD; k0 += 4) {
      v2f a;
      a.x = qa[l16 * 516 + k0 + 2 * half];
      a.y = qa[l16 * 516 + k0 + 2 * half + 1];
      // contiguous pair -> single global_load_b64
      v2f b = *(const v2f*)(bptr + k0 + 2 * half);
      acc = wmma4(a, b, acc);
    }
    for (int r = 0; r < 8; ++r) {
      int row = half * 8 + r;
      sims[row * 132 + wave * 16 + l16] = acc[r];
    }
    __syncthreads();
    if (tid < 16) {  // one thread per query row merges 128 candidates
      for (int j = 0; j < 128; ++j) {
        float v = sims[tid * 132 + j];
        if (v > bestv[KNN - 1]) {
          int p = KNN - 1;
          while (p > 0 && bestv[p - 1] < v) {  // strict: ties keep lower index
            bestv[p] = bestv[p - 1];
            besti[p] = besti[p - 1];
            --p;
          }
          bestv[p] = v;
          besti[p] = c0 + j;
        }
      }
    }
    __syncthreads();
  }

  if (tid < 16) {
    const int q = q0 + tid;
    int* ei = (int*)(dout + (size_t)NN * DD);  // edge_index region (int32)
    for (int j = 0; j < KNN; ++j) {
      ei[(size_t)q * KNN + j] = besti[j];                       // neighbor row
      ei[(size_t)NN * KNN + (size_t)q * KNN + j] = q;           // query row
    }
  }
}

// ---------------- Kernel 4: edge_attr[q] = mean_j out[idx[q][j]] -----------
__global__ __launch_bounds__(256) void edge_attr_kernel(
    const float* __restrict__ out, float* __restrict__ dout) {
  const int q = blockIdx.x;
  const int* ei = (const int*)(dout + (size_t)NN * DD);
  __shared__ int idx[KNN];
  if (threadIdx.x < KNN) idx[threadIdx.x] = ei[(size_t)q * KNN + threadIdx.x];
  __syncthreads();
  float* ea = dout + (size_t)NN * DD + 2 * (size_t)NN * KNN;
  for (int c = threadIdx.x; c < DD; c += 256) {
    float s = 0.0f;
    for (int j = 0; j < KNN; ++j) s += out[(size_t)idx[j] * DD + c];
    ea[(size_t)q * DD + c] = s * (1.0f / KNN);
  }
}

extern "C" void kernel_launch(void* const* d_in, const int* in_sizes, int n_in,
                              void* d_out, int out_size, void* d_ws,
                              size_t ws_size, hipStream_t stream) {
  const float* x  = (const float*)d_in[0];
  const float* W1 = (const float*)d_in[1];
  const float* b1 = (const float*)d_in[2];
  const float* W2 = (const float*)d_in[3];
  const float* b2 = (const float*)d_in[4];
  // d_in[5] is k (==8), compile-time constant here.
  float* out = (float*)d_out;          // first N*D elements = MLP output
  float* xn  = (float*)d_ws;           // 16 MB workspace: normalized rows

  mlp_kernel<<<NN / 16, 256, 0, stream>>>(x, W1, b1, W2, b2, out);
  norm_kernel<<<NN, 256, 0, stream>>>(out, xn);
  knn_kernel<<<NN / 16, 256, 0, stream>>>(xn, out);
  edge_attr_kernel<<<NN, 256, 0, stream>>>(out, out);
}